// PointNetLayer_15229954031657
// MI455X (gfx1250) — compile-verified
//
#include <hip/hip_runtime.h>
#include <hip/hip_bf16.h>
#include <cstddef>

typedef __attribute__((ext_vector_type(2))) float v2f;
typedef __attribute__((ext_vector_type(8))) float v8f;

namespace {
constexpr int   kNPts  = 100000;
constexpr int   kK     = 16;
constexpr int   kCIn   = 64;
constexpr int   kCMid  = 64;
constexpr int   kCOut  = 128;
constexpr float kEps   = 1e-5f;
constexpr float kInvRows = 1.0f / (float)(kNPts * (long long)kK);

constexpr int kAStride = 68;          // 16x64 f32 tile rows padded to 68 floats
constexpr int kWaves   = 8;           // waves (points) per block
constexpr int kBlockT  = kWaves * 32; // 256 threads

// d_ws layout in floats
constexpr int WS_SUM1   = 0;
constexpr int WS_SQ1    = 64;
constexpr int WS_SCALE1 = 128;
constexpr int WS_SHIFT1 = 192;
constexpr int WS_SUM2   = 256;
constexpr int WS_SQ2    = 384;
constexpr int WS_SCALE2 = 512;
constexpr int WS_SHIFT2 = 640;
constexpr int WS_W1T    = 768;                    // [64 n][64 k]
constexpr int WS_W2T    = WS_W1T + kCIn * kCMid;  // [128 n][64 k]
} // namespace

__device__ __forceinline__ v8f wmma_f32x4(v2f a, v2f b, v8f c) {
  // D = A(16x4 f32) * B(4x16 f32) + C(16x16 f32); 8-arg VOP3P form
  return __builtin_amdgcn_wmma_f32_16x16x4_f32(
      /*neg_a=*/false, a, /*neg_b=*/false, b,
      /*c_mod=*/(short)0, c, /*reuse_a=*/false, /*reuse_b=*/false);
}

// Transpose W1 (64x64) and W2 (64x128) into ws so B fragments are contiguous
__global__ void prep_transpose(const float* __restrict__ W1,
                               const float* __restrict__ W2,
                               float* __restrict__ ws) {
  int i = blockIdx.x * blockDim.x + threadIdx.x;
  if (i < kCIn * kCMid) {
    int k = i / kCMid, n = i % kCMid;
    ws[WS_W1T + n * kCIn + k] = W1[i];
  }
  int j = i - kCIn * kCMid;
  if (j >= 0 && j < kCMid * kCOut) {
    int k = j / kCOut, n = j % kCOut;
    ws[WS_W2T + n * kCMid + k] = W2[j];
  }
}

// scale = gamma * rsqrt(var+eps); shift = beta - mean*scale
__global__ void finalize_bn(const float* __restrict__ gamma,
                            const float* __restrict__ beta,
                            float* __restrict__ ws,
                            int c_count, int sum_off, int sq_off,
                            int scale_off, int shift_off) {
  int c = threadIdx.x;
  if (c < c_count) {
    float mean = ws[sum_off + c] * kInvRows;
    float var  = fmaxf(ws[sq_off + c] * kInvRows - mean * mean, 0.0f);
    float rstd = rsqrtf(var + kEps);
    float sc   = gamma[c] * rstd;
    ws[scale_off + c] = sc;
    ws[shift_off + c] = beta[c] - mean * sc;
  }
}

// PHASE 1: stats of layer-1 pre-activation
// PHASE 2: stats of layer-2 pre-activation (uses scale1/shift1)
// PHASE 3: full forward + max-pool output (uses both scale/shift pairs)
template <int PHASE>
__global__ __launch_bounds__(kBlockT) void pointnet_fused(
    const float* __restrict__ feat,
    const int*   __restrict__ refidx,
    const float* __restrict__ b1,
    const float* __restrict__ b2,
    float*       __restrict__ ws,
    float*       __restrict__ out) {
  __shared__ __align__(16) float ldsA[kWaves][16 * kAStride];
  __shared__ float blkstat[2 * kCOut];

  const int lane  = threadIdx.x & 31;
  const int wave  = threadIdx.x >> 5;
  const int point = blockIdx.x * kWaves + wave;
  const int m     = lane & 15;   // row (A) / col (B,D) within tile
  const int khalf = lane >> 4;   // which K-pair this half-wave owns

  float* lds = ldsA[wave];
  const float* W1t = ws + WS_W1T;
  const float* W2t = ws + WS_W2T;

  if (PHASE == 1 || PHASE == 2) {
    const int nstat = (PHASE == 1) ? 2 * kCMid : 2 * kCOut;
    for (int i = threadIdx.x; i < nstat; i += kBlockT) blkstat[i] = 0.0f;
    __syncthreads();
  }

  // ---- gather 16 neighbor feature rows into LDS (A tile, row-major) ----
  for (int r = 0; r < 16; ++r) {
    int src = refidx[point * kK + r];
    v2f v = *(const v2f*)(feat + (size_t)src * kCIn + lane * 2);
    *(v2f*)(lds + r * kAStride + lane * 2) = v;
  }

  // ---- GEMM1: (16x64) @ W1(64x64) via 4 Ntiles x 16 k-steps ----
  v8f acc1[4];
  #pragma unroll
  for (int nt = 0; nt < 4; ++nt) acc1[nt] = (v8f){0, 0, 0, 0, 0, 0, 0, 0};
  #pragma unroll
  for (int s = 0; s < 16; ++s) {
    const int k = 4 * s + 2 * khalf;
    v2f a = *(const v2f*)(lds + m * kAStride + k);
    #pragma unroll
    for (int nt = 0; nt < 4; ++nt) {
      v2f b = *(const v2f*)(W1t + (nt * 16 + m) * kCIn + k);
      acc1[nt] = wmma_f32x4(a, b, acc1[nt]);
    }
  }
  #pragma unroll
  for (int nt = 0; nt < 4; ++nt) {       // + b1
    float bb = b1[nt * 16 + m];
    #pragma unroll
    for (int j = 0; j < 8; ++j) acc1[nt][j] += bb;
  }

  if (PHASE == 1) {
    #pragma unroll
    for (int nt = 0; nt < 4; ++nt) {
      float s = 0.0f, q = 0.0f;
      #pragma unroll
      for (int j = 0; j < 8; ++j) { float y = acc1[nt][j]; s += y; q += y * y; }
      s += __shfl_xor(s, 16, 32);
      q += __shfl_xor(q, 16, 32);
      if (lane < 16) {
        int c = nt * 16 + lane;
        atomicAdd(&blkstat[c], s);
        atomicAdd(&blkstat[kCMid + c], q);
      }
    }
    __syncthreads();
    for (int i = threadIdx.x; i < kCMid; i += kBlockT) {
      atomicAdd(&ws[WS_SUM1 + i], blkstat[i]);
      atomicAdd(&ws[WS_SQ1 + i], blkstat[kCMid + i]);
    }
    return;
  }

  // ---- BN1 + ReLU, restage H1 (D-layout -> row-major LDS -> A-layout) ----
  #pragma unroll
  for (int nt = 0; nt < 4; ++nt) {
    int c = nt * 16 + m;
    float sc = ws[WS_SCALE1 + c];
    float sh = ws[WS_SHIFT1 + c];
    #pragma unroll
    for (int j = 0; j < 8; ++j) {
      float h = fmaxf(acc1[nt][j] * sc + sh, 0.0f);
      lds[(j + 8 * khalf) * kAStride + c] = h;   // row = M, col = channel
    }
  }

  // ---- GEMM2: (16x64) @ W2(64x128) via 8 Ntiles x 16 k-steps ----
  v8f acc2[8];
  #pragma unroll
  for (int nt = 0; nt < 8; ++nt) acc2[nt] = (v8f){0, 0, 0, 0, 0, 0, 0, 0};
  #pragma unroll
  for (int s = 0; s < 16; ++s) {
    const int k = 4 * s + 2 * khalf;
    v2f a = *(const v2f*)(lds + m * kAStride + k);
    #pragma unroll
    for (int nt = 0; nt < 8; ++nt) {
      v2f b = *(const v2f*)(W2t + (nt * 16 + m) * kCMid + k);
      acc2[nt] = wmma_f32x4(a, b, acc2[nt]);
    }
  }
  #pragma unroll
  for (int nt = 0; nt < 8; ++nt) {       // + b2
    float bb = b2[nt * 16 + m];
    #pragma unroll
    for (int j = 0; j < 8; ++j) acc2[nt][j] += bb;
  }

  if (PHASE == 2) {
    #pragma unroll
    for (int nt = 0; nt < 8; ++nt) {
      float s = 0.0f, q = 0.0f;
      #pragma unroll
      for (int j = 0; j < 8; ++j) { float y = acc2[nt][j]; s += y; q += y * y; }
      s += __shfl_xor(s, 16, 32);
      q += __shfl_xor(q, 16, 32);
      if (lane < 16) {
        int c = nt * 16 + lane;
        atomicAdd(&blkstat[c], s);
        atomicAdd(&blkstat[kCOut + c], q);
      }
    }
    __syncthreads();
    for (int i = threadIdx.x; i < kCOut; i += kBlockT) {
      atomicAdd(&ws[WS_SUM2 + i], blkstat[i]);
      atomicAdd(&ws[WS_SQ2 + i], blkstat[kCOut + i]);
    }
    return;
  }

  // ---- PHASE 3: BN2 + ReLU + max over the 16 neighbors ----
  #pragma unroll
  for (int nt = 0; nt < 8; ++nt) {
    int c = nt * 16 + m;
    float sc = ws[WS_SCALE2 + c];
    float sh = ws[WS_SHIFT2 + c];
    float mx = 0.0f;                     // ReLU outputs are >= 0
    #pragma unroll
    for (int j = 0; j < 8; ++j)
      mx = fmaxf(mx, fmaxf(acc2[nt][j] * sc + sh, 0.0f));
    mx = fmaxf(mx, __shfl_xor(mx, 16, 32));  // combine M=0..7 with M=8..15
    if (lane < 16)
      out[(size_t)point * kCOut + c] = mx;
  }
}

extern "C" void kernel_launch(void* const* d_in, const int* in_sizes, int n_in,
                              void* d_out, int out_size, void* d_ws, size_t ws_size,
                              hipStream_t stream) {
  const float* feat   = (const float*)d_in[0];
  const int*   refidx = (const int*)  d_in[1];
  const float* W1     = (const float*)d_in[2];
  const float* b1     = (const float*)d_in[3];
  const float* g1     = (const float*)d_in[4];
  const float* be1    = (const float*)d_in[5];
  const float* W2     = (const float*)d_in[6];
  const float* b2     = (const float*)d_in[7];
  const float* g2     = (const float*)d_in[8];
  const float* be2    = (const float*)d_in[9];
  float* out = (float*)d_out;
  float* ws  = (float*)d_ws;

  // zero the stats/scale region every call (harness does not re-poison)
  hipMemsetAsync(ws, 0, 768 * sizeof(float), stream);

  prep_transpose<<<48, 256, 0, stream>>>(W1, W2, ws);

  dim3 grid(kNPts / kWaves);  // 12500 blocks, 8 waves (points) each
  pointnet_fused<1><<<grid, kBlockT, 0, stream>>>(feat, refidx, b1, b2, ws, out);
  finalize_bn<<<1, kCMid, 0, stream>>>(g1, be1, ws, kCMid,
                                       WS_SUM1, WS_SQ1, WS_SCALE1, WS_SHIFT1);
  pointnet_fused<2><<<grid, kBlockT, 0, stream>>>(feat, refidx, b1, b2, ws, out);
  finalize_bn<<<1, kCOut, 0, stream>>>(g2, be2, ws, kCOut,
                                       WS_SUM2, WS_SQ2, WS_SCALE2, WS_SHIFT2);
  pointnet_fused<3><<<grid, kBlockT, 0, stream>>>(feat, refidx, b1, b2, ws, out);
}